// TypeAwareHAGATLayer_58892591563458
// MI455X (gfx1250) — compile-verified
//
#include <hip/hip_runtime.h>
#include <hip/hip_bf16.h>

// ---------------- problem constants (fixed by the reference) ----------------
constexpr int Nn = 100000;  // nodes per node type
constexpr int Ee = 200000;  // edges per edge type
constexpr int Dd = 64;      // embed dim
constexpr int Hh = 4;       // heads
constexpr int Vv = 50;      // type vocab
constexpr int HD = Hh * Dd; // 256
constexpr float NEG_SLOPE = 0.2f;

typedef __attribute__((ext_vector_type(2))) float v2f;
typedef __attribute__((ext_vector_type(8))) float v8f;

// ---------------------------------------------------------------------------
// fp32 WMMA GEMM, K fixed at 64:  C[m,n] = sum_k A[m,k]*B[k,n] (+ bias[n]).
//
// Block = 256 threads = 8 waves. Block tile: 128(M) x 64(N).
//  * B tile (64x64) staged into LDS once per block via CDNA5 async DMA
//    (global_load_async_to_lds_b128, ASYNCcnt) + s_wait_asynccnt + barrier.
//    The Bs pointer is passed INTO the asm so the array escapes and the
//    "memory" clobber makes the later ds_load reads of Bs real.
//  * Each wave computes a 16x64 strip: A fragments (16 x v2f, contiguous b64
//    loads) loaded once, reused across 4 N-subtiles; inner loop interleaves
//    4 independent accumulator chains of V_WMMA_F32_16X16X4_F32 (64 WMMAs).
//
// VGPR layouts per CDNA5 ISA 7.12.2 (32-bit data):
//   A 16x4:  lanes 0-15 hold M=0..15 with K=0(V0),K=1(V1); lanes 16-31 K=2,3
//   B 4x16:  V0 = row K=0 (lanes0-15) / K=2 (lanes16-31); V1 = K=1 / K=3
//   C 16x16: V_i: lanes0-15 -> M=i, lanes16-31 -> M=i+8; N = lane&15
// ---------------------------------------------------------------------------
constexpr int KDIM = 64;
constexpr int KPAD = 68;            // padded LDS row stride (floats)
constexpr int NSUB = 4;             // 4 x 16 = 64-wide N block per wave
constexpr int BN   = 16 * NSUB;     // 64
constexpr int WAVES = 8;
constexpr int BM   = 16 * WAVES;    // 128 rows per block

__global__ __launch_bounds__(256)
void wmma_gemm_f32_lds(const float* __restrict__ A, int lda, long sAz,
                       const float* __restrict__ B, int ldb, long sBz,
                       float* __restrict__ C, int ldc, long sCz,
                       int Mtotal, const float* __restrict__ bias) {
  __shared__ float Bs[KDIM][KPAD];

  const int tid  = threadIdx.x;
  const int lane = tid & 31;
  const int wave = tid >> 5;
  const int z    = blockIdx.z;
  A += (long)z * sAz;  B += (long)z * sBz;  C += (long)z * sCz;
  const int n0 = blockIdx.y * BN;

  // ---- stage B tile (64x64) into LDS via async DMA (16B per lane-op) ----
  {
    float* bsp = &Bs[0][0];   // escapes into the asm -> LDS writes are visible
#pragma unroll
    for (int rep = 0; rep < (KDIM * BN) / (256 * 4); ++rep) {   // 4 reps
      const int l = (tid + rep * 256) * 4;          // element index in tile
      const int k = l >> 6;                         // row (BN == 64)
      const int n = l & 63;                         // col, float4 aligned
      const unsigned lds_off = (unsigned)(k * KPAD + n) * 4u;
      const unsigned long long ga =
          (unsigned long long)(B + (long)k * ldb + n0 + n);
      asm volatile("global_load_async_to_lds_b128 %0, %1, off"
                   :: "v"(lds_off), "v"(ga), "r"(bsp) : "memory");
    }
    asm volatile("s_wait_asynccnt 0x0" ::: "memory");
  }
  __syncthreads();

  // ---- per-wave 16x64 strip ----
  const int mn   = lane & 15;      // A row / B,C column within 16-tile
  const int half = lane >> 4;      // K-half selector
  const int m0   = blockIdx.x * BM + wave * 16;

  const int arow = min(m0 + mn, Mtotal - 1);      // clamp (stores are guarded)
  const float* Ar = A + (long)arow * lda + (half << 1);

  v2f a[16];
#pragma unroll
  for (int kk = 0; kk < 16; ++kk)
    a[kk] = *(const v2f*)(Ar + (kk << 2));        // contiguous b64 loads

  v8f c[NSUB];
#pragma unroll
  for (int j = 0; j < NSUB; ++j) {
    const float bv = bias ? bias[n0 + j * 16 + mn] : 0.0f;
#pragma unroll
    for (int i = 0; i < 8; ++i) c[j][i] = bv;
  }

#pragma unroll
  for (int kk = 0; kk < 16; ++kk) {
    const int ka = (kk << 2) + (half << 1);
#pragma unroll
    for (int j = 0; j < NSUB; ++j) {              // 4 independent chains
      v2f b;
      b[0] = Bs[ka + 0][j * 16 + mn];
      b[1] = Bs[ka + 1][j * 16 + mn];
      c[j] = __builtin_amdgcn_wmma_f32_16x16x4_f32(false, a[kk], false, b,
                                                   (short)0, c[j], false, false);
    }
  }

#pragma unroll
  for (int j = 0; j < NSUB; ++j)
#pragma unroll
    for (int i = 0; i < 8; ++i) {
      const int row = m0 + (half << 3) + i;
      if (row < Mtotal)
        C[(long)row * ldc + n0 + j * 16 + mn] = c[j][i];
    }
}

// ---------------------------------------------------------------------------
// small helpers
// ---------------------------------------------------------------------------
__device__ __forceinline__ unsigned f32_order(float f) {
  unsigned u = __float_as_uint(f);
  return (u & 0x80000000u) ? ~u : (u | 0x80000000u);   // order-preserving map
}
__device__ __forceinline__ float order_f32(unsigned u) {
  return (u & 0x80000000u) ? __uint_as_float(u & 0x7fffffffu)
                           : __uint_as_float(~u);
}
__device__ __forceinline__ float dot64(const float* __restrict__ a,
                                       const float* __restrict__ b) {
  float s = 0.f;
#pragma unroll
  for (int i = 0; i < 16; ++i) {
    const float4 av = ((const float4*)a)[i];
    const float4 bv = ((const float4*)b)[i];
    s += av.x * bv.x + av.y * bv.y + av.z * bv.z + av.w * bv.w;
  }
  return s;
}

__global__ void fill_u32(unsigned* __restrict__ p, unsigned v, long n) {
  long i = (long)blockIdx.x * blockDim.x + threadIdx.x;
  if (i < n) p[i] = v;
}

// wtop[h*64+f] = sum_d Wdst[h,f,d]*adst[h,d]; wbot uses rows 64..127
__global__ void fold_wdst_kernel(const float* __restrict__ Wdst,
                                 const float* __restrict__ adst,
                                 float* __restrict__ wtop,
                                 float* __restrict__ wbot) {
  const int t = threadIdx.x;                // 256 threads: (h,f)
  const int h = t >> 6, f = t & 63;
  wtop[t] = dot64(Wdst + (long)h * 128 * 64 + (long)f * 64,        adst + h * 64);
  wbot[t] = dot64(Wdst + (long)h * 128 * 64 + (long)(64 + f) * 64, adst + h * 64);
}

// Tsrc[v,h,d] = sum_f type_emb[v,f] * Wsrc[h, 64+f, d]
__global__ void type_table_kernel(const float* __restrict__ type_emb,
                                  const float* __restrict__ Wsrc,
                                  float* __restrict__ Tsrc) {
  const int v = blockIdx.x;
  const int h = threadIdx.x >> 6, d = threadIdx.x & 63;
  const float* te = type_emb + (long)v * 64;
  const float* w  = Wsrc + (long)h * 128 * 64 + 64 * 64 + d;  // rows 64..127
  float s = 0.f;
#pragma unroll 8
  for (int f = 0; f < 64; ++f) s += te[f] * w[(long)f * 64];
  Tsrc[(long)v * 256 + h * 64 + d] = s;
}

// per-node scalars: out[n*4+h] = M[n,h,:]·vec[h,:]
__global__ void node_scalar_kernel(const float* __restrict__ M, int ldm,
                                   const float* __restrict__ vec,
                                   float* __restrict__ out, int n_total) {
  int i = blockIdx.x * blockDim.x + threadIdx.x;   // i = n*4+h
  if (i >= n_total) return;
  int n = i >> 2, h = i & 3;
  out[i] = dot64(M + (long)n * ldm + h * 64, vec + h * 64);
}

// per-type scalars: tsrc[v,h]=Tsrc[v,h,:]·asrc[h];  tdst[v,h]=type_emb[v]·wbot[h]
__global__ void type_scalar_kernel(const float* __restrict__ Tsrc,
                                   const float* __restrict__ asrc,
                                   const float* __restrict__ type_emb,
                                   const float* __restrict__ wbot,
                                   float* __restrict__ tsrc,
                                   float* __restrict__ tdst) {
  int i = blockIdx.x * blockDim.x + threadIdx.x;   // i = v*4+h
  if (i >= Vv * Hh) return;
  int v = i >> 2, h = i & 3;
  tsrc[i] = dot64(Tsrc + (long)v * 256 + h * 64, asrc + h * 64);
  tdst[i] = dot64(type_emb + (long)v * 64, wbot + h * 64);
}

// ---------------------------------------------------------------------------
// edge phase
// ---------------------------------------------------------------------------
__global__ void edge_logit_kernel(const int* __restrict__ ei,
                                  const int* __restrict__ et,
                                  const float* __restrict__ ssrc,
                                  const float* __restrict__ sdst,
                                  const float* __restrict__ tsrc,
                                  const float* __restrict__ tdst,
                                  float* __restrict__ lg,
                                  unsigned* __restrict__ smax) {
  int e = blockIdx.x * blockDim.x + threadIdx.x;
  if (e >= Ee) return;
  const int s = ei[e], d = ei[Ee + e], t = et[e];
#pragma unroll
  for (int h = 0; h < 4; ++h) {
    float v = ssrc[s * 4 + h] + tsrc[t * 4 + h] + sdst[d * 4 + h] + tdst[t * 4 + h];
    v = v > 0.f ? v : NEG_SLOPE * v;            // leaky_relu
    lg[(long)e * 4 + h] = v;
    atomicMax(&smax[d * 4 + h], f32_order(v));  // segment max
  }
}

__global__ void edge_exp_kernel(const int* __restrict__ ei,
                                float* __restrict__ lg,     // in: logit, out: exp
                                const unsigned* __restrict__ smax,
                                float* __restrict__ den) {
  int e = blockIdx.x * blockDim.x + threadIdx.x;
  if (e >= Ee) return;
  const int d = ei[Ee + e];
#pragma unroll
  for (int h = 0; h < 4; ++h) {
    float mx = order_f32(smax[d * 4 + h]);
    float ex = __expf(lg[(long)e * 4 + h] - mx);
    lg[(long)e * 4 + h] = ex;
    atomicAdd(&den[d * 4 + h], ex);
  }
}

// one 256-thread block per edge: thread (h,d) scatters alpha*(Psrc+Tsrc)
__global__ void edge_agg_kernel(const int* __restrict__ ei,
                                const int* __restrict__ et,
                                const float* __restrict__ ex,
                                const float* __restrict__ den,
                                const float* __restrict__ Psrc,
                                const float* __restrict__ Tsrc,
                                float* __restrict__ out) {
  const int e = blockIdx.x;
  const int h = threadIdx.x >> 6, d = threadIdx.x & 63;
  const int s = ei[e], dn = ei[Ee + e], t = et[e];
  const float alpha = ex[(long)e * 4 + h] / (den[dn * 4 + h] + 1e-16f);
  const float val = alpha * (Psrc[(long)s * 256 + h * 64 + d] +
                             Tsrc[(long)t * 256 + h * 64 + d]);
  atomicAdd(&out[(long)dn * 256 + h * 64 + d], val);
}

__global__ void elu_kernel(float* __restrict__ p, long n) {
  long i = (long)blockIdx.x * blockDim.x + threadIdx.x;
  if (i >= n) return;
  float x = p[i];
  p[i] = x > 0.f ? x : (__expf(x) - 1.f);
}

// ---------------------------------------------------------------------------
extern "C" void kernel_launch(void* const* d_in, const int* in_sizes, int n_in,
                              void* d_out, int out_size, void* d_ws, size_t ws_size,
                              hipStream_t stream) {
  const float* x_a   = (const float*)d_in[0];
  const float* x_b   = (const float*)d_in[1];
  const float* W_a   = (const float*)d_in[2];
  const float* b_a   = (const float*)d_in[3];
  const float* W_b   = (const float*)d_in[4];
  const float* b_b   = (const float*)d_in[5];
  const float* temb  = (const float*)d_in[6];
  const float* Wsrc_ab = (const float*)d_in[7];
  const float* Wdst_ab = (const float*)d_in[8];
  const float* asrc_ab = (const float*)d_in[9];
  const float* adst_ab = (const float*)d_in[10];
  const float* Wsrc_ba = (const float*)d_in[11];
  const float* Wdst_ba = (const float*)d_in[12];
  const float* asrc_ba = (const float*)d_in[13];
  const float* adst_ba = (const float*)d_in[14];
  const int* ei_ab = (const int*)d_in[15];
  const int* et_ab = (const int*)d_in[16];
  const int* ei_ba = (const int*)d_in[17];
  const int* et_ba = (const int*)d_in[18];

  float* out_a = (float*)d_out;                       // [N,H,D] type-a nodes
  float* out_b = (float*)d_out + (long)Nn * HD;       // [N,H,D] type-b nodes

  // ---- workspace carve-up (floats) ----
  float* ws = (float*)d_ws;
  long off = 0;
  auto alloc = [&](long nf) { float* p = ws + off; off += nf; return p; };
  float* h_a      = alloc((long)Nn * HD);
  float* h_b      = alloc((long)Nn * HD);
  float* Psrc_ab  = alloc((long)Nn * HD);
  float* Psrc_ba  = alloc((long)Nn * HD);
  float* Tsrc_ab  = alloc((long)Vv * HD);
  float* Tsrc_ba  = alloc((long)Vv * HD);
  float* wtop_ab  = alloc(HD);  float* wbot_ab = alloc(HD);
  float* wtop_ba  = alloc(HD);  float* wbot_ba = alloc(HD);
  float* ssrc_ab  = alloc((long)Nn * Hh);
  float* sdst_ab  = alloc((long)Nn * Hh);
  float* ssrc_ba  = alloc((long)Nn * Hh);
  float* sdst_ba  = alloc((long)Nn * Hh);
  float* tsrc_ab  = alloc(Vv * Hh); float* tdst_ab = alloc(Vv * Hh);
  float* tsrc_ba  = alloc(Vv * Hh); float* tdst_ba = alloc(Vv * Hh);
  unsigned* smax_ab = (unsigned*)alloc((long)Nn * Hh);
  unsigned* smax_ba = (unsigned*)alloc((long)Nn * Hh);
  float* den_ab   = alloc((long)Nn * Hh);
  float* den_ba   = alloc((long)Nn * Hh);
  float* ex_ab    = alloc((long)Ee * Hh);
  float* ex_ba    = alloc((long)Ee * Hh);

  const unsigned ORD_NEG_INF = 0x007FFFFFu;  // f32_order(-inf)
  auto blocks = [](long n, int b) { return (unsigned)((n + b - 1) / b); };

  // ---- init ----
  fill_u32<<<blocks((long)2 * Nn * HD, 256), 256, 0, stream>>>((unsigned*)d_out, 0u, (long)2 * Nn * HD);
  fill_u32<<<blocks((long)Nn * Hh, 256), 256, 0, stream>>>((unsigned*)den_ab, 0u, (long)Nn * Hh);
  fill_u32<<<blocks((long)Nn * Hh, 256), 256, 0, stream>>>((unsigned*)den_ba, 0u, (long)Nn * Hh);
  fill_u32<<<blocks((long)Nn * Hh, 256), 256, 0, stream>>>(smax_ab, ORD_NEG_INF, (long)Nn * Hh);
  fill_u32<<<blocks((long)Nn * Hh, 256), 256, 0, stream>>>(smax_ba, ORD_NEG_INF, (long)Nn * Hh);

  // ---- node GEMMs: h = x @ W + b   [N,64]x[64,256], K=64 ----
  const unsigned gm = blocks(Nn, BM);               // 782 row-blocks
  dim3 gA(gm, HD / BN, 1);                          // (782, 4)
  wmma_gemm_f32_lds<<<gA, 256, 0, stream>>>(x_a, Dd, 0, W_a, HD, 0,
                                            h_a, HD, 0, Nn, b_a);
  wmma_gemm_f32_lds<<<gA, 256, 0, stream>>>(x_b, Dd, 0, W_b, HD, 0,
                                            h_b, HD, 0, Nn, b_b);

  // ---- per-head source projections: P[n,h,:] = h[n,h,:] @ Wsrc_top[h] ----
  dim3 gP(gm, 1, Hh);                               // z = head, N=64
  wmma_gemm_f32_lds<<<gP, 256, 0, stream>>>(h_a, HD, Dd, Wsrc_ab, Dd, (long)2 * Dd * Dd,
                                            Psrc_ab, HD, Dd, Nn, nullptr);
  wmma_gemm_f32_lds<<<gP, 256, 0, stream>>>(h_b, HD, Dd, Wsrc_ba, Dd, (long)2 * Dd * Dd,
                                            Psrc_ba, HD, Dd, Nn, nullptr);

  // ---- tiny precomputes ----
  fold_wdst_kernel<<<1, 256, 0, stream>>>(Wdst_ab, adst_ab, wtop_ab, wbot_ab);
  fold_wdst_kernel<<<1, 256, 0, stream>>>(Wdst_ba, adst_ba, wtop_ba, wbot_ba);
  type_table_kernel<<<Vv, 256, 0, stream>>>(temb, Wsrc_ab, Tsrc_ab);
  type_table_kernel<<<Vv, 256, 0, stream>>>(temb, Wsrc_ba, Tsrc_ba);

  node_scalar_kernel<<<blocks((long)Nn * Hh, 256), 256, 0, stream>>>(Psrc_ab, HD, asrc_ab, ssrc_ab, Nn * Hh);
  node_scalar_kernel<<<blocks((long)Nn * Hh, 256), 256, 0, stream>>>(h_b,     HD, wtop_ab, sdst_ab, Nn * Hh);
  node_scalar_kernel<<<blocks((long)Nn * Hh, 256), 256, 0, stream>>>(Psrc_ba, HD, asrc_ba, ssrc_ba, Nn * Hh);
  node_scalar_kernel<<<blocks((long)Nn * Hh, 256), 256, 0, stream>>>(h_a,     HD, wtop_ba, sdst_ba, Nn * Hh);

  type_scalar_kernel<<<1, 256, 0, stream>>>(Tsrc_ab, asrc_ab, temb, wbot_ab, tsrc_ab, tdst_ab);
  type_scalar_kernel<<<1, 256, 0, stream>>>(Tsrc_ba, asrc_ba, temb, wbot_ba, tsrc_ba, tdst_ba);

  // ---- edge phase (ab: dst are b-nodes; ba: dst are a-nodes) ----
  const unsigned eg = blocks(Ee, 256);
  edge_logit_kernel<<<eg, 256, 0, stream>>>(ei_ab, et_ab, ssrc_ab, sdst_ab, tsrc_ab, tdst_ab, ex_ab, smax_ab);
  edge_logit_kernel<<<eg, 256, 0, stream>>>(ei_ba, et_ba, ssrc_ba, sdst_ba, tsrc_ba, tdst_ba, ex_ba, smax_ba);
  edge_exp_kernel<<<eg, 256, 0, stream>>>(ei_ab, ex_ab, smax_ab, den_ab);
  edge_exp_kernel<<<eg, 256, 0, stream>>>(ei_ba, ex_ba, smax_ba, den_ba);
  edge_agg_kernel<<<Ee, 256, 0, stream>>>(ei_ab, et_ab, ex_ab, den_ab, Psrc_ab, Tsrc_ab, out_b);
  edge_agg_kernel<<<Ee, 256, 0, stream>>>(ei_ba, et_ba, ex_ba, den_ba, Psrc_ba, Tsrc_ba, out_a);

  // ---- ELU finalize ----
  elu_kernel<<<blocks((long)2 * Nn * HD, 256), 256, 0, stream>>>((float*)d_out, (long)2 * Nn * HD);
  (void)in_sizes; (void)n_in; (void)out_size; (void)ws_size;
}